// HaarDeconv2D_9732395893262
// MI455X (gfx1250) — compile-verified
//
#include <hip/hip_runtime.h>
#include <hip/hip_bf16.h>
#include <stdint.h>

// Haar vertical deconv == row interleave with Haar reconstruction arithmetic.
// Inputs:  lp, dt  [B=16, C=3, H=512, W=1024] fp32  (96 MB each)
// Output:  out     [B, C, 2H, W]              fp32  (192 MB)
// Memory-bound: 384 MB total traffic -> ~16.5 us floor at 23.3 TB/s.
//
// CDNA5 path: per-lane GLOBAL_LOAD_ASYNC_TO_LDS_B128 (ASYNCcnt DMA) stages
// both input tiles into LDS; s_wait_asynccnt 0; ds_load_b128 back; VALU does
// the 0.5*(s +/- d) reconstruction; non-temporal B128 stores stream out.

typedef float f4 __attribute__((ext_vector_type(4)));

__global__ __launch_bounds__(256) void haar_deconv_interleave(
    const float* __restrict__ lp, const float* __restrict__ dt,
    float* __restrict__ out, unsigned total_units)
{
    // 8 KB LDS: [0..1023] = lp tile, [1024..2047] = dt tile (floats)
    __shared__ float smem[2048];

    const unsigned tid = threadIdx.x;
    const unsigned u   = blockIdx.x * 256u + tid;   // float4 unit id
    if (u >= total_units) return;

    const unsigned row = u >> 8;            // W=1024 -> 256 float4 per row
    const unsigned col = (u & 255u) << 2;   // float column within row

    // Global source addresses (64-bit, per lane)
    const unsigned long long g_lp = (unsigned long long)(uintptr_t)(lp + (size_t)u * 4);
    const unsigned long long g_dt = (unsigned long long)(uintptr_t)(dt + (size_t)u * 4);

    // LDS byte offsets. Low 32 bits of a generic pointer to LDS are the
    // wave-relative LDS offset (ISA 10.2 aperture rules), which is exactly
    // what the async instruction's VDST expects (an LDS byte address in the VDST register).
    const unsigned lds_base = (unsigned)(uintptr_t)(&smem[0]);
    const unsigned off_lp   = lds_base + tid * 16u;
    const unsigned off_dt   = off_lp + 4096u;

    // Async DMA: global -> LDS, 128b per lane, tracked by ASYNCcnt.
    asm volatile("global_load_async_to_lds_b128 %0, %1, off"
                 :: "v"(off_lp), "v"(g_lp) : "memory");
    asm volatile("global_load_async_to_lds_b128 %0, %1, off"
                 :: "v"(off_dt), "v"(g_dt) : "memory");
    // Each wave consumes only LDS bytes its own lanes wrote -> wave-local wait,
    // no workgroup barrier required.
    asm volatile("s_wait_asynccnt 0" ::: "memory");

    const f4 a = *(const f4*)(&smem[tid * 4u]);          // low_pass
    const f4 b = *(const f4*)(&smem[1024u + tid * 4u]);  // detail

    // Faithful Haar reconstruction arithmetic (matches reference rounding):
    const f4 s    = a + b;                 // lp + detail   (kernel L path)
    const f4 d    = a - b;                 // lp - detail   (kernel D path)
    const f4 even = 0.5f * s + 0.5f * d;   // == lp (up to rounding)
    const f4 odd  = 0.5f * s - 0.5f * d;   // == detail (up to rounding)

    // Interleave rows: out[..., 2*row, col] / out[..., 2*row+1, col]
    float* oe = out + (size_t)row * 2048u + col;
    __builtin_nontemporal_store(even, (f4*)oe);            // th:NT streaming store
    __builtin_nontemporal_store(odd,  (f4*)(oe + 1024));
}

extern "C" void kernel_launch(void* const* d_in, const int* in_sizes, int n_in,
                              void* d_out, int out_size, void* d_ws, size_t ws_size,
                              hipStream_t stream) {
    const float* lp  = (const float*)d_in[0];
    const float* dt  = (const float*)d_in[1];
    float*       out = (float*)d_out;

    const unsigned n_elems = (unsigned)in_sizes[0];      // 25,165,824
    const unsigned units   = n_elems / 4u;               // float4 units: 6,291,456
    const unsigned blocks  = (units + 255u) / 256u;      // 24,576 blocks x 8 waves

    haar_deconv_interleave<<<blocks, 256, 0, stream>>>(lp, dt, out, units);
}